// NodeEmbedding_12575664243034
// MI455X (gfx1250) — compile-verified
//
#include <hip/hip_runtime.h>
#include <hip/hip_bf16.h>

#define NUM_NODE  50000
#define WINDOW    96
#define TEM_DIM   64
#define SPA_DIM   64
#define BATCHSZ   8
#define N_EDGE    800000
#define OUT_DIM   128   // TEM_DIM + SPA_DIM

typedef __attribute__((ext_vector_type(2))) float v2f;
typedef __attribute__((ext_vector_type(8))) float v8f;

// ---------------------------------------------------------------------------
// Temporal branch: out[b,n,0:64] = 0.9*temb[n,:] + 0.1*(x[b,n,:] @ fc_w^T + fc_b)
// One wave computes a 16(row)x64(col) tile via V_WMMA_F32_16X16X4_F32,
// accumulating K=96 in 24 steps of 4.
// ---------------------------------------------------------------------------
__global__ __launch_bounds__(256) void tem_gemm_kernel(
    const float* __restrict__ x,      // [B*N, 96]
    const float* __restrict__ temb,   // [N, 64]
    const float* __restrict__ fc_w,   // [64, 96]  (out, in) row-major
    const float* __restrict__ fc_b,   // [64]
    float* __restrict__ out)          // [B*N, 128]
{
    const int lane   = threadIdx.x & 31;
    const int laneLo = lane & 15;
    const int laneHi = lane >> 4;
    const int tile   = blockIdx.x * 8 + (threadIdx.x >> 5);   // 25000 tiles exactly
    const int row0   = tile * 16;                             // flattened b*N + node
    if (row0 >= BATCHSZ * NUM_NODE) return;                   // wave-uniform

    // A: lane holds x[row0+laneLo, k0 + 2*laneHi + {0,1}]  (8B aligned v2f)
    const float* aptr  = x    + (size_t)(row0 + laneLo) * WINDOW + 2 * laneHi;
    // B: lane holds fc_w^T[k, n] = fc_w[n*96 + k], n = t*16 + laneLo
    const float* bbase = fc_w + (size_t)laneLo * WINDOW + 2 * laneHi;

    v8f acc[4] = {};
    for (int k = 0; k < WINDOW; k += 4) {
        v2f a = *(const v2f*)(aptr + k);
#pragma unroll
        for (int t = 0; t < 4; ++t) {
            v2f b = *(const v2f*)(bbase + (size_t)t * 16 * WINDOW + k);
            acc[t] = __builtin_amdgcn_wmma_f32_16x16x4_f32(
                false, a, false, b, (short)0, acc[t], false, false);
        }
    }

    // Epilogue: D VGPR r -> (M = r + 8*laneHi, N = t*16 + laneLo)
    const int nodeBase = row0 % NUM_NODE;   // tiles never straddle a batch (50000 % 16 == 0)
#pragma unroll
    for (int t = 0; t < 4; ++t) {
#pragma unroll
        for (int r = 0; r < 8; ++r) {
            const int m    = r + 8 * laneHi;
            const int n    = t * 16 + laneLo;
            const int row  = row0 + m;
            const int node = nodeBase + m;
            const float v  = 0.9f * temb[(size_t)node * TEM_DIM + n]
                           + 0.1f * (acc[t][r] + fc_b[n]);
            out[(size_t)row * OUT_DIM + n] = v;
        }
    }
}

// ---------------------------------------------------------------------------
// GCN linear: xw = spatial_embedding @ gcn_w^T   [50000,64] x [64,64]
// ---------------------------------------------------------------------------
__global__ __launch_bounds__(256) void gcn_xw_kernel(
    const float* __restrict__ semb,   // [N, 64]
    const float* __restrict__ gcn_w,  // [64, 64] (out, in) row-major
    float* __restrict__ xw)           // [N, 64]
{
    const int lane   = threadIdx.x & 31;
    const int laneLo = lane & 15;
    const int laneHi = lane >> 4;
    const int tile   = blockIdx.x * 8 + (threadIdx.x >> 5);
    if (tile * 16 >= NUM_NODE) return;                        // wave-uniform
    const int row0 = tile * 16;

    const float* aptr  = semb  + (size_t)(row0 + laneLo) * SPA_DIM + 2 * laneHi;
    const float* bbase = gcn_w + (size_t)laneLo * SPA_DIM + 2 * laneHi;

    v8f acc[4] = {};
    for (int k = 0; k < SPA_DIM; k += 4) {
        v2f a = *(const v2f*)(aptr + k);
#pragma unroll
        for (int t = 0; t < 4; ++t) {
            v2f b = *(const v2f*)(bbase + (size_t)t * 16 * SPA_DIM + k);
            acc[t] = __builtin_amdgcn_wmma_f32_16x16x4_f32(
                false, a, false, b, (short)0, acc[t], false, false);
        }
    }
#pragma unroll
    for (int t = 0; t < 4; ++t) {
#pragma unroll
        for (int r = 0; r < 8; ++r) {
            const int m = r + 8 * laneHi;
            const int n = t * 16 + laneLo;
            xw[(size_t)(row0 + m) * SPA_DIM + n] = acc[t][r];
        }
    }
}

// ---------------------------------------------------------------------------
// GCN scatter path helpers
// ---------------------------------------------------------------------------
__global__ void init_kernel(float* __restrict__ deg, float* __restrict__ spa_acc) {
    const int i = blockIdx.x * blockDim.x + threadIdx.x;
    if (i < NUM_NODE * SPA_DIM) spa_acc[i] = 0.0f;
    if (i < NUM_NODE)           deg[i]     = 1.0f;   // self-loop contribution
}

__global__ void deg_kernel(const long long* __restrict__ ei, float* __restrict__ deg) {
    const int e = blockIdx.x * blockDim.x + threadIdx.x;
    if (e < N_EDGE) {
        const int c = (int)ei[(size_t)N_EDGE + e];   // col = target
        atomicAdd(&deg[c], 1.0f);
    }
}

__global__ void dinv_kernel(float* __restrict__ deg) {
    const int i = blockIdx.x * blockDim.x + threadIdx.x;
    if (i < NUM_NODE) deg[i] = rsqrtf(deg[i]);       // deg >= 1 (self-loops)
}

// One wave per edge; each lane handles 2 of the 64 feature dims.
// xw (12.8MB) and spa_acc (12.8MB) both live in the 192MB L2.
__global__ __launch_bounds__(256) void scatter_kernel(
    const long long* __restrict__ ei, const float* __restrict__ dinv,
    const float* __restrict__ xw, float* __restrict__ spa_acc)
{
    const int e = blockIdx.x * 8 + (threadIdx.x >> 5);
    if (e >= N_EDGE) return;                         // wave-uniform
    const int lane = threadIdx.x & 31;
    const int r = (int)ei[e];                        // row = source
    const int c = (int)ei[(size_t)N_EDGE + e];       // col = target
    const float norm = dinv[r] * dinv[c];
    v2f v = *(const v2f*)(xw + (size_t)r * SPA_DIM + lane * 2);
    float* dst = spa_acc + (size_t)c * SPA_DIM + lane * 2;
    atomicAdd(dst,     norm * v[0]);
    atomicAdd(dst + 1, norm * v[1]);
}

// spa = acc + dinv^2 * xw (self-loop) + bias; broadcast to all 8 batches.
__global__ void finalize_kernel(
    const float* __restrict__ spa_acc, const float* __restrict__ dinv,
    const float* __restrict__ xw, const float* __restrict__ gcn_b,
    float* __restrict__ out)
{
    const int i = blockIdx.x * blockDim.x + threadIdx.x;
    if (i >= NUM_NODE * SPA_DIM) return;
    const int n = i / SPA_DIM;
    const int d = i - n * SPA_DIM;
    const float di  = dinv[n];
    const float val = spa_acc[i] + di * di * xw[i] + gcn_b[d];
#pragma unroll
    for (int b = 0; b < BATCHSZ; ++b)
        out[((size_t)b * NUM_NODE + n) * OUT_DIM + TEM_DIM + d] = val;
}

// ---------------------------------------------------------------------------
extern "C" void kernel_launch(void* const* d_in, const int* in_sizes, int n_in,
                              void* d_out, int out_size, void* d_ws, size_t ws_size,
                              hipStream_t stream) {
    (void)in_sizes; (void)n_in; (void)out_size; (void)ws_size;
    // setup_inputs() dict order:
    const float*     x     = (const float*)d_in[0];  // [8,50000,96]
    const float*     temb  = (const float*)d_in[1];  // [50000,64]
    const float*     semb  = (const float*)d_in[2];  // [50000,64]
    const float*     fc_w  = (const float*)d_in[3];  // [64,96]
    const float*     fc_b  = (const float*)d_in[4];  // [64]
    const float*     gcn_w = (const float*)d_in[5];  // [64,64]
    const float*     gcn_b = (const float*)d_in[6];  // [64]
    const long long* ei    = (const long long*)d_in[7]; // [2,800000] int64
    float* out = (float*)d_out;                      // [8,50000,128]

    // Workspace layout (all fp32, 16B-aligned regions):
    float* deg = (float*)d_ws;                       // 50000 (padded to 50176)
    float* xw  = deg + 50176;                        // 50000*64
    float* spa = xw + (size_t)NUM_NODE * SPA_DIM;    // 50000*64   (~25.8 MB total)

    // GCN branch
    init_kernel   <<<(NUM_NODE * SPA_DIM + 255) / 256, 256, 0, stream>>>(deg, spa);
    deg_kernel    <<<(N_EDGE + 255) / 256,             256, 0, stream>>>(ei, deg);
    dinv_kernel   <<<(NUM_NODE + 255) / 256,           256, 0, stream>>>(deg);
    gcn_xw_kernel <<<(NUM_NODE / 16 + 7) / 8,          256, 0, stream>>>(semb, gcn_w, xw);
    scatter_kernel<<<(N_EDGE + 7) / 8,                 256, 0, stream>>>(ei, deg, xw, spa);
    finalize_kernel<<<(NUM_NODE * SPA_DIM + 255) / 256,256, 0, stream>>>(spa, deg, xw, gcn_b, out);

    // Temporal branch (largest kernel: 25000 wave-tiles, 8 waves/block)
    tem_gemm_kernel<<<(BATCHSZ * NUM_NODE / 16) / 8,   256, 0, stream>>>(x, temb, fc_w, fc_b, out);
}